// CosineLoss_10170482557111
// MI455X (gfx1250) — compile-verified
//
#include <hip/hip_runtime.h>
#include <hip/hip_bf16.h>

typedef float v2f __attribute__((ext_vector_type(2)));
typedef float v4f __attribute__((ext_vector_type(4)));
typedef float v8f __attribute__((ext_vector_type(8)));

#define MARGIN 0.5f
#define EPS_F 1e-8f

// One wave (32 lanes) owns 16 rows. Per 32-column group each lane streams
// float4 (b128) chunks of x and y, folds the elementwise products
// (x*y, x*x, y*y) into its two A-tile slots, then three
// V_WMMA_F32_16X16X4_F32 against an all-ones B fold the 16x4 A-tiles into
// per-row running sums (B=1 => D[M][*] = row-sum of A row M).
__global__ __launch_bounds__(256) void cosloss_main_kernel(
    const float* __restrict__ x, const float* __restrict__ y,
    const int* __restrict__ p, float* __restrict__ partial, int D)
{
    const int lane = threadIdx.x & 31;
    const int wid  = threadIdx.x >> 5;            // 8 waves per block
    const int waveGlobal = blockIdx.x * 8 + wid;  // 16 rows per wave
    const int rowBase = waveGlobal * 16;
    const int h = lane >> 4;                      // lane half
    const int m = lane & 15;                      // row within the 16-row tile

    // lane m handles columns k+8c+0..3, lane m+16 handles k+8c+4..7
    const size_t rowOff = (size_t)(rowBase + m) * (size_t)D + (size_t)(4 * h);
    const float* xp = x + rowOff;
    const float* yp = y + rowOff;

    v8f cdot = {}; v8f cxx = {}; v8f cyy = {};
    v2f ones; ones.x = 1.0f; ones.y = 1.0f;       // B = all-ones => row-sum

#pragma unroll 2
    for (int k = 0; k < D; k += 32) {
        v2f adot = {}; v2f axx = {}; v2f ayy = {};
#pragma unroll
        for (int c = 0; c < 4; ++c) {
            v4f ax = __builtin_nontemporal_load((const v4f*)(xp + k + 8 * c));
            v4f ay = __builtin_nontemporal_load((const v4f*)(yp + k + 8 * c));
            adot.x += ax.x * ay.x;  adot.y += ax.y * ay.y;
            adot.x += ax.z * ay.z;  adot.y += ax.w * ay.w;
            axx.x  += ax.x * ax.x;  axx.y  += ax.y * ax.y;
            axx.x  += ax.z * ax.z;  axx.y  += ax.w * ax.w;
            ayy.x  += ay.x * ay.x;  ayy.y  += ay.y * ay.y;
            ayy.x  += ay.z * ay.z;  ayy.y  += ay.w * ay.w;
        }
        // (neg_a, A, neg_b, B, c_mod, C, reuse_a, reuse_b)
        cdot = __builtin_amdgcn_wmma_f32_16x16x4_f32(false, adot, false, ones,
                                                     (short)0, cdot, false, false);
        cxx  = __builtin_amdgcn_wmma_f32_16x16x4_f32(false, axx,  false, ones,
                                                     (short)0, cxx,  false, false);
        cyy  = __builtin_amdgcn_wmma_f32_16x16x4_f32(false, ayy,  false, ones,
                                                     (short)0, cyy,  false, false);
    }

    // C/D layout: VGPR r, lanes 0-15 -> row M=r ; lanes 16-31 -> row M=r+8.
    // Every N column is identical, so lane 0 and lane 16 hold all 16 row-sums.
    __shared__ float red[16];
    if (m == 0) {
        float acc = 0.0f;
        const int rbase = rowBase + 8 * h;
#pragma unroll
        for (int r = 0; r < 8; ++r) {
            float dot = cdot[r];
            float xx  = cxx[r];
            float yy  = cyy[r];
            float denom = fmaxf(__builtin_sqrtf(xx) * __builtin_sqrtf(yy), EPS_F);
            float d = dot / denom;
            float pp = (p[rbase + r] == 1) ? (1.0f - d)
                                           : fmaxf(0.0f, d - MARGIN);
            acc += pp;
        }
        red[wid * 2 + h] = acc;
    }
    __syncthreads();
    if (threadIdx.x == 0) {
        float s = 0.0f;
#pragma unroll
        for (int i = 0; i < 16; ++i) s += red[i];
        partial[blockIdx.x] = s;
    }
}

// Deterministic fixed-order tree reduction of the per-block partials.
__global__ __launch_bounds__(256) void cosloss_reduce_kernel(
    const float* __restrict__ partial, float* __restrict__ out, int n)
{
    __shared__ float red[256];
    float s = 0.0f;
    for (int i = threadIdx.x; i < n; i += 256) s += partial[i];
    red[threadIdx.x] = s;
    __syncthreads();
    for (int off = 128; off > 0; off >>= 1) {
        if ((int)threadIdx.x < off) red[threadIdx.x] += red[threadIdx.x + off];
        __syncthreads();
    }
    if (threadIdx.x == 0) out[0] = red[0];
}

extern "C" void kernel_launch(void* const* d_in, const int* in_sizes, int n_in,
                              void* d_out, int out_size, void* d_ws, size_t ws_size,
                              hipStream_t stream) {
    const float* x = (const float*)d_in[0];
    const float* y = (const float*)d_in[1];
    const int*   p = (const int*)d_in[2];

    const int N = in_sizes[2];            // 32768 rows
    const int D = in_sizes[0] / N;        // 1024 columns

    float* partial = (float*)d_ws;        // one float per block

    const int rowsPerBlock = 128;         // 8 waves * 16 rows
    const int nBlocks = N / rowsPerBlock; // 256

    cosloss_main_kernel<<<nBlocks, 256, 0, stream>>>(x, y, p, partial, D);
    cosloss_reduce_kernel<<<1, 256, 0, stream>>>(partial, (float*)d_out, nBlocks);
}